// TorchReshapedGatherEinsum_24902220382296
// MI455X (gfx1250) — compile-verified
//
#include <hip/hip_runtime.h>
#include <stdint.h>

// CDNA5 / gfx1250: wave32, f32 WMMA 16x16x4 (exact-precision, HBM-bound kernel),
// double-buffered LDS staging via GLOBAL_LOAD_ASYNC_TO_LDS_B128 (ASYNCcnt).

typedef float v2f __attribute__((ext_vector_type(2)));
typedef float v8f __attribute__((ext_vector_type(8)));

// Problem constants (match reference)
constexpr int Bc = 4;
constexpr int T  = 4096;
constexpr int I  = 1024;
constexpr int J  = 1024;
constexpr int E  = 16;
constexpr int K  = 512;

// Tiling
constexpr int TM  = 128;  // output rows (gathered K-slots) per block
constexpr int TN  = 128;  // output cols (J) per block
constexpr int TK  = 16;   // I-chunk per pipeline stage
constexpr int LDA = 20;   // A row stride in floats (80 B = 16B multiple)
constexpr int LDB = 132;  // B row stride in floats (528 B = 16B multiple, bank-skewed)

constexpr int ABUF = TM * LDA;   // floats per A buffer (2560)
constexpr int BBUF = TK * LDB;   // floats per B buffer (2112)
constexpr int NCHUNK = I / TK;   // 64

__device__ __forceinline__ void async_copy_b128(uint32_t lds_byte_off, const float* gptr) {
    // per-lane: copies 16 bytes from global address gptr to LDS byte offset lds_byte_off ; tracked by ASYNCcnt
    asm volatile("global_load_async_to_lds_b128 %0, %1, off"
                 :: "v"(lds_byte_off), "v"(gptr)
                 : "memory");
}

__device__ __forceinline__ void wait_async0() {
    asm volatile("s_wait_asynccnt 0x0" ::: "memory");
}

__global__ __launch_bounds__(256)
void gather_expert_gemm_f32(const float* __restrict__ X,
                            const int*   __restrict__ ind,
                            const float* __restrict__ W,
                            float*       __restrict__ Y)
{
    __shared__ __align__(16) float lds_A[2 * ABUF];   // 20480 B
    __shared__ __align__(16) float lds_B[2 * BBUF];   // 16896 B
    __shared__ int s_idx[TM];                         //   512 B

    const int tid  = threadIdx.x;
    const int lane = tid & 31;
    const int wave = tid >> 5;
    const int hi   = lane >> 4;    // half-wave select per WMMA f32 layouts
    const int lrow = lane & 15;

    // block -> (b, e, m-tile, n-tile)
    constexpr int tilesPerBE = (K / TM) * (J / TN);   // 4 * 8 = 32
    const int be   = blockIdx.x / tilesPerBE;
    const int tile = blockIdx.x % tilesPerBE;
    const int b  = be / E;
    const int e  = be % E;
    const int mt = tile / (J / TN);
    const int nt = tile % (J / TN);
    const int m0 = mt * TM;
    const int n0 = nt * TN;

    // wave -> 64x32 sub-tile inside the 128x128 block tile (2x4 wave grid)
    const int wm = (wave >> 2) * 64;
    const int wn = (wave & 3)  * 32;

    if (tid < TM) {
        s_idx[tid] = ind[((size_t)b * E + e) * K + m0 + tid];
    }
    __syncthreads();   // s_idx visible before first async issue

    const float* Xb = X + (size_t)b * T * I;
    const float* We = W + (size_t)e * I * J + n0;

    // Async-stage one TK chunk (A: TM x TK gathered X rows, B: TK x TN of W) into buffer `buf`.
    auto issue_chunk = [&](int kk, int buf) {
        // A: TM*TK/4 = 512 b128 transfers -> 2 per thread
        #pragma unroll
        for (int i = 0; i < (TM * TK / 4) / 256; ++i) {
            int lin = tid + 256 * i;
            int row = lin / (TK / 4);          // 4 float4 per row
            int c4  = (lin % (TK / 4)) * 4;
            const float* g = Xb + (size_t)s_idx[row] * I + kk + c4;
            uint32_t dst = (uint32_t)(size_t)&lds_A[buf * ABUF + row * LDA + c4];
            async_copy_b128(dst, g);
        }
        // B: TK*TN/4 = 512 b128 transfers -> 2 per thread
        #pragma unroll
        for (int i = 0; i < (TK * TN / 4) / 256; ++i) {
            int lin = tid + 256 * i;
            int row = lin >> 5;                // 32 float4 per row
            int c4  = (lin & 31) << 2;
            const float* g = We + (size_t)(kk + row) * J + c4;
            uint32_t dst = (uint32_t)(size_t)&lds_B[buf * BBUF + row * LDB + c4];
            async_copy_b128(dst, g);
        }
    };

    const v8f vzero = {0.f, 0.f, 0.f, 0.f, 0.f, 0.f, 0.f, 0.f};
    v8f acc[4][2];
    #pragma unroll
    for (int mi = 0; mi < 4; ++mi)
        #pragma unroll
        for (int ni = 0; ni < 2; ++ni)
            acc[mi][ni] = vzero;

    issue_chunk(0, 0);

    int cur = 0;
    for (int c = 0; c < NCHUNK; ++c) {
        wait_async0();      // my wave's async copies for chunk c done
        __syncthreads();    // => every wave's copies done; LDS buffer `cur` ready

        if (c + 1 < NCHUNK)
            issue_chunk((c + 1) * TK, cur ^ 1);   // overlap next chunk with compute

        const int aOff = cur * ABUF;
        const int bOff = cur * BBUF;

        #pragma unroll
        for (int k = 0; k < TK; k += 4) {
            // A fragments: 16x4 f32; lanes 0-15 -> K=k,k+1 ; lanes 16-31 -> K=k+2,k+3
            v2f a[4];
            #pragma unroll
            for (int mi = 0; mi < 4; ++mi)
                a[mi] = *(const v2f*)(&lds_A[aOff + (wm + mi * 16 + lrow) * LDA + k + 2 * hi]);

            // B fragments: 4x16 f32; VGPR0: row k (lo half) / k+2 (hi half); VGPR1: +1
            v2f bf[2];
            #pragma unroll
            for (int ni = 0; ni < 2; ++ni) {
                int col = wn + ni * 16 + lrow;
                bf[ni].x = lds_B[bOff + (k + 2 * hi)     * LDB + col];
                bf[ni].y = lds_B[bOff + (k + 2 * hi + 1) * LDB + col];
            }

            #pragma unroll
            for (int mi = 0; mi < 4; ++mi)
                #pragma unroll
                for (int ni = 0; ni < 2; ++ni)
                    acc[mi][ni] = __builtin_amdgcn_wmma_f32_16x16x4_f32(
                        /*neg_a=*/false, a[mi],
                        /*neg_b=*/false, bf[ni],
                        /*c_mod=*/(short)0, acc[mi][ni],
                        /*reuse_a=*/false, /*reuse_b=*/false);
        }
        cur ^= 1;
    }

    // Epilogue: C/D layout — VGPR r: lanes 0-15 -> M=r, lanes 16-31 -> M=r+8; N=lane&15
    float* Yt = Y + ((size_t)b * E + e) * (size_t)K * J;
    #pragma unroll
    for (int mi = 0; mi < 4; ++mi)
        #pragma unroll
        for (int ni = 0; ni < 2; ++ni)
            #pragma unroll
            for (int r = 0; r < 8; ++r) {
                int row = m0 + wm + mi * 16 + r + 8 * hi;
                int col = n0 + wn + ni * 16 + lrow;
                Yt[(size_t)row * J + col] = acc[mi][ni][r];
            }
}

extern "C" void kernel_launch(void* const* d_in, const int* in_sizes, int n_in,
                              void* d_out, int out_size, void* d_ws, size_t ws_size,
                              hipStream_t stream) {
    (void)in_sizes; (void)n_in; (void)out_size; (void)d_ws; (void)ws_size;
    const float* X  = (const float*)d_in[0];
    const int*   nd = (const int*)  d_in[1];
    const float* W  = (const float*)d_in[2];
    float*       Y  = (float*)d_out;

    constexpr int tilesPerBE = (K / TM) * (J / TN);       // 32
    const int grid = Bc * E * tilesPerBE;                 // 2048 blocks
    gather_expert_gemm_f32<<<grid, 256, 0, stream>>>(X, nd, W, Y);
}